// AFNB_1503238554169
// MI455X (gfx1250) — compile-verified
//
#include <hip/hip_runtime.h>
#include <math.h>

typedef __attribute__((ext_vector_type(2))) float v2f;
typedef __attribute__((ext_vector_type(8))) float v8f;

#define BN_EPS 1e-5f

// ---------------------------------------------------------------------------
// PSP adaptive-avg-pool (sizes 1,3,6,8 -> 110 bins) via LDS integral image.
// One block per (channel, batch) plane of 64x64.
// out[b*outBatchStride + bin*strideBin + c*strideC]; bins >=110 written as 0.
// ---------------------------------------------------------------------------
__global__ void psp_pool_kernel(const float* __restrict__ X, float* __restrict__ out,
                                int C, int strideBin, int strideC,
                                long outBatchStride, int padBins) {
    __shared__ float sm[64 * 65];
    const int c = blockIdx.x;
    const int b = blockIdx.y;
    const float* x = X + ((size_t)b * C + c) * 4096;
    const int tid = threadIdx.x;

    for (int i = tid; i < 4096; i += blockDim.x) {
        int r = i >> 6, cc = i & 63;
        sm[r * 65 + cc] = x[i];
    }
    __syncthreads();
    if (tid < 64) {               // prefix along rows
        float s = 0.f;
        for (int j = 0; j < 64; ++j) { s += sm[tid * 65 + j]; sm[tid * 65 + j] = s; }
    }
    __syncthreads();
    if (tid < 64) {               // prefix along cols -> inclusive integral image
        float s = 0.f;
        for (int r = 0; r < 64; ++r) { s += sm[r * 65 + tid]; sm[r * 65 + tid] = s; }
    }
    __syncthreads();
    if (tid < padBins) {
        float* o = out + (size_t)b * outBatchStride + (size_t)tid * strideBin
                       + (size_t)c * strideC;
        if (tid < 110) {
            int s, idx;
            if (tid == 0)       { s = 1; idx = 0; }
            else if (tid < 10)  { s = 3; idx = tid - 1; }
            else if (tid < 46)  { s = 6; idx = tid - 10; }
            else                { s = 8; idx = tid - 46; }
            int i = idx / s, j = idx % s;
            int hs = (i * 64) / s, he = ((i + 1) * 64 + s - 1) / s;
            int ws = (j * 64) / s, we = ((j + 1) * 64 + s - 1) / s;
            float v = sm[(he - 1) * 65 + (we - 1)];
            if (hs > 0)           v -= sm[(hs - 1) * 65 + (we - 1)];
            if (ws > 0)           v -= sm[(he - 1) * 65 + (ws - 1)];
            if (hs > 0 && ws > 0) v += sm[(hs - 1) * 65 + (ws - 1)];
            *o = v / (float)((he - hs) * (we - ws));
        } else {
            *o = 0.f;  // zero padding so padded GEMM tiles contribute nothing
        }
    }
}

// ---------------------------------------------------------------------------
// valueP[b,o,j] = sum_c wv[o,c]*Plow[b,c,j] + bv[o]   (tiny: 0.23 GFLOP total)
// Plow layout [B,1024,112]; valueP layout [B,256,112], cols 110/111 = 0.
// ---------------------------------------------------------------------------
__global__ void value_pool_gemm(const float* __restrict__ wv, const float* __restrict__ bv,
                                const float* __restrict__ Plow, float* __restrict__ valueP) {
    int t = blockIdx.x * blockDim.x + threadIdx.x;   // 0 .. 256*112-1
    int b = blockIdx.y;
    if (t >= 256 * 112) return;
    int o = t / 112, j = t % 112;
    float acc = 0.f;
    if (j < 110) {
        const float* p = Plow + (size_t)b * 1024 * 112 + j;
        const float* w = wv + (size_t)o * 1024;
        for (int c = 0; c < 1024; ++c) acc += w[c] * p[(size_t)c * 112];
        acc += bv[o];
    }
    valueP[(size_t)b * 256 * 112 + (size_t)o * 112 + j] = acc;
}

// ---------------------------------------------------------------------------
// In-place column softmax over 110 bins of layout [B][128][N]; rows 110/111 -> 0.
// ---------------------------------------------------------------------------
__global__ void softmax_kernel(float* __restrict__ sa, int N) {
    int n = blockIdx.x * blockDim.x + threadIdx.x;
    int b = blockIdx.y;
    float* p = sa + (size_t)b * 128 * N + n;
    float mx = -3.4e38f;
    for (int m = 0; m < 110; ++m) mx = fmaxf(mx, p[(size_t)m * N]);
    float sum = 0.f;
    for (int m = 0; m < 110; ++m) sum += __expf(p[(size_t)m * N] - mx);
    float inv = 1.f / sum;
    for (int m = 0; m < 110; ++m) p[(size_t)m * N] = __expf(p[(size_t)m * N] - mx) * inv;
    p[(size_t)110 * N] = 0.f;
    p[(size_t)111 * N] = 0.f;
}

// ---------------------------------------------------------------------------
// fp32 WMMA GEMM: C[M,N] = epi(scale * A[M,K] x B[K,N]).
// One wave (32 threads) per 64x64 C tile; V_WMMA_F32_16X16X4_F32, K step 4.
// EPI: 0=none, 1=bias+BN+ReLU, 2=bias, 3=bias+BN.
// SPLITB: B rows [0,ksplit) from B0, [ksplit,K) from B1 (virtual concat).
// ---------------------------------------------------------------------------
template <int EPI, bool SPLITB>
__global__ void __launch_bounds__(32) wmma_gemm_f32(
    const float* __restrict__ A, long aBatchStride,
    const float* __restrict__ B0, long b0BatchStride,
    const float* __restrict__ B1, long b1BatchStride, int ksplit,
    float* __restrict__ C, long cBatchStride,
    int M, int N, int K, float scale,
    const float* __restrict__ bias,
    const float* __restrict__ gamma, const float* __restrict__ beta,
    const float* __restrict__ mean, const float* __restrict__ var) {
    const int b  = blockIdx.z;
    const int n0 = blockIdx.x * 64;
    const int m0 = blockIdx.y * 64;
    const int lane = threadIdx.x;
    const int lo = lane & 15;    // M (A) / N (B,C) position within tile
    const int hi = lane >> 4;    // K-half select for A/B, M+8 select for C

    const float* Ab  = A  + (size_t)b * aBatchStride;
    const float* B0b = B0 + (size_t)b * b0BatchStride;
    const float* B1b = SPLITB ? (B1 + (size_t)b * b1BatchStride) : nullptr;
    float*       Cb  = C  + (size_t)b * cBatchStride;

    v8f acc[4][4] = {};

    for (int k = 0; k < K; k += 4) {
        const float* Bsrc;
        int kb;
        if (SPLITB && k >= ksplit) { Bsrc = B1b; kb = k - ksplit; }
        else                       { Bsrc = B0b; kb = k; }
        const float* brow = Bsrc + (size_t)(kb + hi * 2) * N;

        v2f afrag[4];
#pragma unroll
        for (int mi = 0; mi < 4; ++mi) {
            const float* ap = Ab + (size_t)(m0 + mi * 16 + lo) * K + (k + hi * 2);
            afrag[mi] = *(const v2f*)ap;       // A[M=lo, K=k+2*hi .. +1]
        }
        v2f bfrag[4];
#pragma unroll
        for (int ni = 0; ni < 4; ++ni) {
            const float* bp = brow + (n0 + ni * 16 + lo);
            bfrag[ni].x = bp[0];               // B[K=k+2*hi,   N=lo]
            bfrag[ni].y = bp[N];               // B[K=k+2*hi+1, N=lo]
        }
#pragma unroll
        for (int mi = 0; mi < 4; ++mi)
#pragma unroll
            for (int ni = 0; ni < 4; ++ni)
                acc[mi][ni] = __builtin_amdgcn_wmma_f32_16x16x4_f32(
                    false, afrag[mi], false, bfrag[ni], (short)0, acc[mi][ni],
                    false, false);
    }

#pragma unroll
    for (int mi = 0; mi < 4; ++mi) {
#pragma unroll
        for (int r = 0; r < 8; ++r) {
            const int row = m0 + mi * 16 + hi * 8 + r;
            float mulv = 1.f, addv = 0.f;
            if (EPI == 1 || EPI == 3) {
                float inv = gamma[row] * rsqrtf(var[row] + BN_EPS);
                mulv = inv;
                addv = beta[row] - mean[row] * inv + bias[row] * inv;
            } else if (EPI == 2) {
                addv = bias[row];
            }
#pragma unroll
            for (int ni = 0; ni < 4; ++ni) {
                float v = acc[mi][ni][r] * scale;
                v = v * mulv + addv;
                if (EPI == 1) v = fmaxf(v, 0.f);
                Cb[(size_t)row * N + (n0 + ni * 16 + lo)] = v;
            }
        }
    }
}

// ---------------------------------------------------------------------------
extern "C" void kernel_launch(void* const* d_in, const int* in_sizes, int n_in,
                              void* d_out, int out_size, void* d_ws, size_t ws_size,
                              hipStream_t stream) {
    const float* low  = (const float*)d_in[0];   // [4,1024,64,64]
    const float* high = (const float*)d_in[1];   // [4,512,64,64]
    const float* wv   = (const float*)d_in[2];
    const float* bv   = (const float*)d_in[3];
    const float* wq   = (const float*)d_in[4];
    const float* bq   = (const float*)d_in[5];
    const float* qg   = (const float*)d_in[6];
    const float* qb   = (const float*)d_in[7];
    const float* qm   = (const float*)d_in[8];
    const float* qv   = (const float*)d_in[9];
    const float* wk   = (const float*)d_in[10];
    const float* bk   = (const float*)d_in[11];
    const float* kg   = (const float*)d_in[12];
    const float* kb   = (const float*)d_in[13];
    const float* km   = (const float*)d_in[14];
    const float* kvr  = (const float*)d_in[15];
    const float* wW   = (const float*)d_in[16];
    const float* bW   = (const float*)d_in[17];
    const float* wo   = (const float*)d_in[18];
    const float* bo   = (const float*)d_in[19];
    const float* og   = (const float*)d_in[20];
    const float* ob   = (const float*)d_in[21];
    const float* om   = (const float*)d_in[22];
    const float* ov   = (const float*)d_in[23];
    float* out = (float*)d_out;

    const int Bn = 4, N = 4096;
    float* ws = (float*)d_ws;
    size_t off = 0;
    float* Qfull = ws + off; off += (size_t)Bn * 256 * N;      // later reused as ctxCN
    float* Kfull = ws + off; off += (size_t)Bn * 256 * N;
    float* ctx2  = ws + off; off += (size_t)Bn * 512 * N;
    float* simT  = ws + off; off += (size_t)Bn * 128 * N;      // softmax in-place
    float* Plow  = ws + off; off += (size_t)Bn * 1024 * 112;
    float* valP  = ws + off; off += (size_t)Bn * 256 * 112;
    float* keyPT = ws + off; off += (size_t)Bn * 128 * 256;
    (void)ws_size; (void)in_sizes; (void)n_in; (void)out_size;

    // 1. Plow = psp(low)  [B,1024,112]  (pooling commutes with the value conv)
    psp_pool_kernel<<<dim3(1024, Bn), 128, 0, stream>>>(
        low, Plow, 1024, /*strideBin*/1, /*strideC*/112, (long)1024 * 112, 112);

    // 2. valueP = wv x Plow + bv   [B,256,112]
    value_pool_gemm<<<dim3(112, Bn), 256, 0, stream>>>(wv, bv, Plow, valP);

    // 3. Qfull = relu(bn(wq x high + bq))   M=256 K=512
    wmma_gemm_f32<1, false><<<dim3(64, 4, Bn), 32, 0, stream>>>(
        wq, 0, high, (long)512 * N, nullptr, 0, 0,
        Qfull, (long)256 * N, 256, N, 512, 1.f, bq, qg, qb, qm, qv);

    // 4. Kfull = relu(bn(wk x low + bk))    M=256 K=1024
    wmma_gemm_f32<1, false><<<dim3(64, 4, Bn), 32, 0, stream>>>(
        wk, 0, low, (long)1024 * N, nullptr, 0, 0,
        Kfull, (long)256 * N, 256, N, 1024, 1.f, bk, kg, kb, km, kvr);

    // 5. keyPT = psp(Kfull)^T  [B,128,256], bins 110..127 zero
    psp_pool_kernel<<<dim3(256, Bn), 128, 0, stream>>>(
        Kfull, keyPT, 256, /*strideBin*/256, /*strideC*/1, (long)128 * 256, 128);

    // 6. simT = (keyPT x Qfull) / 16   M=128 K=256
    wmma_gemm_f32<0, false><<<dim3(64, 2, Bn), 32, 0, stream>>>(
        keyPT, (long)128 * 256, Qfull, (long)256 * N, nullptr, 0, 0,
        simT, (long)128 * N, 128, N, 256, 0.0625f,
        nullptr, nullptr, nullptr, nullptr, nullptr);

    // 7. attn = softmax(sim) over 110 bins, in place; pad rows zeroed
    softmax_kernel<<<dim3(16, Bn), 256, 0, stream>>>(simT, N);

    // 8. ctxCN = valueP x attnT   M=256 K=112  (reuse Qfull storage)
    float* ctxCN = Qfull;
    wmma_gemm_f32<0, false><<<dim3(64, 4, Bn), 32, 0, stream>>>(
        valP, (long)256 * 112, simT, (long)128 * N, nullptr, 0, 0,
        ctxCN, (long)256 * N, 256, N, 112, 1.f,
        nullptr, nullptr, nullptr, nullptr, nullptr);

    // 9. ctx2 = wW x ctxCN + bW   M=512 K=256
    wmma_gemm_f32<2, false><<<dim3(64, 8, Bn), 32, 0, stream>>>(
        wW, 0, ctxCN, (long)256 * N, nullptr, 0, 0,
        ctx2, (long)512 * N, 512, N, 256, 1.f,
        bW, nullptr, nullptr, nullptr, nullptr);

    // 10. out = bn(wo x [ctx2; high] + bo)   M=512 K=1024 (virtual concat)
    wmma_gemm_f32<3, true><<<dim3(64, 8, Bn), 32, 0, stream>>>(
        wo, 0, ctx2, (long)512 * N, high, (long)512 * N, 512,
        out, (long)512 * N, 512, N, 1024, 1.f, bo, og, ob, om, ov);
}